// MultiQueryAttention_37744172597487
// MI455X (gfx1250) — compile-verified
//
#include <hip/hip_runtime.h>
#include <math.h>

typedef float v2f __attribute__((ext_vector_type(2)));
typedef float v8f __attribute__((ext_vector_type(8)));
typedef float f4  __attribute__((ext_vector_type(4)));
typedef int   v4i __attribute__((ext_vector_type(4)));
typedef unsigned int u32x4 __attribute__((ext_vector_type(4)));
typedef int i32x8 __attribute__((ext_vector_type(8)));
typedef int i32x4 __attribute__((ext_vector_type(4)));

#define WMMA_F32(a, b, c) \
  __builtin_amdgcn_wmma_f32_16x16x4_f32(false, (a), false, (b), (short)0, (c), false, false)

// ---------------------------------------------------------------------------
// Async global->LDS staging (CDNA5 GLOBAL_LOAD_ASYNC_TO_LDS_B128, ASYNCcnt).
// ---------------------------------------------------------------------------
typedef __attribute__((address_space(1))) v4i g_v4i;
typedef __attribute__((address_space(3))) v4i l_v4i;
typedef __attribute__((address_space(3))) float l_f32;

#if __has_builtin(__builtin_amdgcn_global_load_async_to_lds_b128)
#define ASYNC_LDS 1
__device__ __forceinline__ void async_cp16(const float* g, float* l) {
  __builtin_amdgcn_global_load_async_to_lds_b128((g_v4i*)g, (l_v4i*)l, 0, 0);
}
#else
#define ASYNC_LDS 0
__device__ __forceinline__ void async_cp16(const float* g, float* l) {
  *(f4*)l = *(const f4*)g;
}
#endif

__device__ __forceinline__ void async_wait0() {
#if ASYNC_LDS
#if __has_builtin(__builtin_amdgcn_s_wait_asynccnt)
  __builtin_amdgcn_s_wait_asynccnt(0);
#else
  asm volatile("s_wait_asynccnt 0x0" ::: "memory");
#endif
#endif
}

#if __has_builtin(__builtin_amdgcn_tensor_load_to_lds)
#define HAS_TDM 1
#else
#define HAS_TDM 0
#endif

// ---------------------------------------------------------------------------
// C[M,N] = X[M,K] @ W[N,K]^T   (torch Linear convention)
// 256 threads = 8 waves (2 along M x 4 along N); block tile 128x128, BK=32.
// Double-buffered LDS with async global->LDS staging; one barrier per K-step.
// ---------------------------------------------------------------------------
__global__ __launch_bounds__(256)
void gemm_xwt_f32(const float* __restrict__ X, const float* __restrict__ W,
                  float* __restrict__ C, int M, int N, int K) {
  constexpr int BM = 128, BN = 128, BK = 32, LS = BK + 4; // 36
  __shared__ float As[2][BM * LS];
  __shared__ float Bs[2][BN * LS];

  const int tid  = threadIdx.x;
  const int wave = tid >> 5;
  const int lane = tid & 31;
  const int half = lane >> 4;
  const int l16  = lane & 15;
  const int wm = (wave >> 2) * 64;
  const int wn = (wave & 3) * 32;
  const size_t rowA0 = (size_t)blockIdx.y * BM;
  const size_t rowB0 = (size_t)blockIdx.x * BN;

  const int lrow = tid >> 1;
  const int lcol = (tid & 1) * 16;

  auto stage = [&](int buf, int k0) {
    const float* ga = X + (rowA0 + lrow) * (size_t)K + k0 + lcol;
    const float* gb = W + (rowB0 + lrow) * (size_t)K + k0 + lcol;
    float* la = &As[buf][lrow * LS + lcol];
    float* lb = &Bs[buf][lrow * LS + lcol];
#pragma unroll
    for (int ii = 0; ii < 4; ++ii) {
      async_cp16(ga + ii * 4, la + ii * 4);
      async_cp16(gb + ii * 4, lb + ii * 4);
    }
  };

  v8f acc[4][2] = {};

  const int nk = K / BK;
  stage(0, 0);

  for (int t = 0; t < nk; ++t) {
    async_wait0();
    __syncthreads();
    if (t + 1 < nk) stage((t + 1) & 1, (t + 1) * BK);

    const float* as = As[t & 1];
    const float* bs = Bs[t & 1];
#pragma unroll
    for (int c = 0; c < 8; ++c) {
      const int kk = c * 4 + half * 2;
      v2f af[4], bf[2];
#pragma unroll
      for (int i = 0; i < 4; ++i) af[i] = *(const v2f*)&as[(wm + i * 16 + l16) * LS + kk];
#pragma unroll
      for (int j = 0; j < 2; ++j) bf[j] = *(const v2f*)&bs[(wn + j * 16 + l16) * LS + kk];
#pragma unroll
      for (int i = 0; i < 4; ++i)
#pragma unroll
        for (int j = 0; j < 2; ++j)
          acc[i][j] = WMMA_F32(af[i], bf[j], acc[i][j]);
    }
  }

#pragma unroll
  for (int i = 0; i < 4; ++i)
#pragma unroll
    for (int j = 0; j < 2; ++j)
#pragma unroll
      for (int v = 0; v < 8; ++v) {
        size_t r  = rowA0 + wm + i * 16 + v + 8 * half;
        size_t cg = rowB0 + wn + j * 16 + l16;
        C[r * (size_t)N + cg] = acc[i][j][v];
      }
}

// ---------------------------------------------------------------------------
// Rotary embedding: rows x (nh*128), each thread handles one (i, i+64) pair.
// ---------------------------------------------------------------------------
__global__ void rope_f32(const float* __restrict__ in, float* __restrict__ out,
                         int rows, int nh, int S) {
  int idx = blockIdx.x * blockDim.x + threadIdx.x;
  int total = rows * nh * 64;
  if (idx >= total) return;
  int i   = idx & 63;
  int t   = idx >> 6;
  int h   = t % nh;
  int row = t / nh;
  int pos = row % S;
  float inv_freq = expf(-(float)i * (logf(10000.0f) / 64.0f));
  float ang = (float)pos * inv_freq;
  float sv, cv;
  sincosf(ang, &sv, &cv);
  size_t base = (size_t)row * ((size_t)nh * 128) + (size_t)h * 128 + i;
  float x1 = in[base];
  float x2 = in[base + 64];
  out[base]      = x1 * cv - x2 * sv;
  out[base + 64] = x2 * cv + x1 * sv;
}

// ---------------------------------------------------------------------------
// Flash MQA: block = (b, head, 128 query rows); 8 waves x 16 rows each.
// K tile staged by the Tensor Data Mover (TENSOR_LOAD_TO_LDS) with LDS padding
// (pad_interval=128 DW, pad_amount=4 DW -> row stride 132 floats), issued by
// wave 0 while the other waves transpose V. Falls back to async loads.
// ---------------------------------------------------------------------------
__global__ __launch_bounds__(256)
void mqa_attn_f32(const float* __restrict__ Q, const float* __restrict__ Kc,
                  const float* __restrict__ Vc, float* __restrict__ Ctx,
                  int B, int S) {
  constexpr int LS = 132;
  __shared__ float kbuf[128 * LS]; // K tile, reused as P staging
  __shared__ float vbuf[128 * LS]; // V^T tile

  const int tiles_m = S >> 7;
  const int blk = blockIdx.x;
  const int mt = blk % tiles_m;
  const int bh = blk / tiles_m;
  const int h = bh & 15;
  const int b = bh >> 4;
  const int tid = threadIdx.x, wave = tid >> 5, lane = tid & 31;
  const int half = lane >> 4, l16 = lane & 15;
  const int q0 = mt * 128 + wave * 16;

  // Q fragments, pre-scaled by 1/sqrt(128)
  v2f qf[32];
  {
    const float* qrow = Q + ((size_t)(b * S + q0 + l16)) * 2048 + h * 128 + half * 2;
    const float sc = 0.08838834764831845f;
#pragma unroll
    for (int c = 0; c < 32; ++c) {
      v2f t = *(const v2f*)(qrow + c * 4);
      qf[c].x = t.x * sc;
      qf[c].y = t.y * sc;
    }
  }

  v8f o_acc[8] = {};
  float m_i[8], l_i[8];
#pragma unroll
  for (int v = 0; v < 8; ++v) { m_i[v] = -INFINITY; l_i[v] = 0.f; }

  const int lrow = tid >> 1;
  const int lcol = (tid & 1) * 64;

#if HAS_TDM
  const unsigned int lds_kbuf = (unsigned int)(size_t)(l_f32*)kbuf;
#endif

  for (int kt = 0; kt < S; kt += 128) {
#if HAS_TDM
    // --- K tile via Tensor Data Mover (one issue per workgroup, wave 0) ---
    if (wave == 0) {
      unsigned long long ga = (unsigned long long)(size_t)(Kc + ((size_t)(b * S + kt)) * 128);
      // D# group0: count=1 | lds_addr | global_addr[56:0] | type=2
      u32x4 g0;
      g0.x = 1u;                                     // count=1, user mode, no gather
      g0.y = lds_kbuf;                               // lds_addr
      g0.z = (unsigned int)ga;                       // global_addr[31:0]
      g0.w = (unsigned int)(ga >> 32) | (2u << 30);  // global_addr[56:32] | type=2
      // D# group1: data_size=4B, pad_enable, pad_interval=128DW(code 6),
      // pad_amount=4DW(code 3); tensor 128 x S (f32), tile 128x128, stride 128
      i32x8 g1;
      g1[0] = (int)((2u << 16) | (1u << 20) | (6u << 22) | (3u << 25));
      g1[1] = (int)(128u << 16);   // tensor_dim0[15:0] = 128  (bits 79:48 low half)
      g1[2] = (int)(((unsigned)S & 0xFFFFu) << 16); // tensor_dim1[15:0] (bits 111:80 low half)
      g1[3] = (int)(128u << 16);   // tile_dim0 = 128 (bits 127:112)
      g1[4] = 128;                 // tile_dim1 = 128 (bits 143:128)
      g1[5] = 128;                 // tensor_dim0_stride[31:0] = 128 elements
      g1[6] = 0;                   // dim0_stride[47:32] | tensor_dim1_stride[15:0]
      g1[7] = 0;                   // tensor_dim1_stride[47:16]
      i32x4 gz4 = {0, 0, 0, 0};            // groups 2/3: unused (<=2D tile)
      i32x8 gz8 = {0, 0, 0, 0, 0, 0, 0, 0};
      __builtin_amdgcn_tensor_load_to_lds(g0, g1, gz4, gz4, gz8, 0);
    }
#else
    {
      const float* gk = Kc + ((size_t)(b * S + kt + lrow)) * 128 + lcol;
      float* dk = &kbuf[lrow * LS + lcol];
#pragma unroll
      for (int ii = 0; ii < 16; ++ii)
        async_cp16(gk + ii * 4, dk + ii * 4);
    }
#endif
    // --- V tile transposed via registers (overlaps the K DMA) ---
    {
      const float* gv = Vc + ((size_t)(b * S + kt + lrow)) * 128 + lcol;
#pragma unroll
      for (int ii = 0; ii < 16; ++ii) {
        f4 vv = *(const f4*)(gv + ii * 4);
        vbuf[(lcol + ii * 4 + 0) * LS + lrow] = vv.x;
        vbuf[(lcol + ii * 4 + 1) * LS + lrow] = vv.y;
        vbuf[(lcol + ii * 4 + 2) * LS + lrow] = vv.z;
        vbuf[(lcol + ii * 4 + 3) * LS + lrow] = vv.w;
      }
    }
#if HAS_TDM
    if (wave == 0) __builtin_amdgcn_s_wait_tensorcnt(0);
#else
    async_wait0();
#endif
    __syncthreads();

    // scores: S = Q @ K^T (16 x 128 per wave)
    v8f s_acc[8] = {};
#pragma unroll 4
    for (int c = 0; c < 32; ++c) {
      const int kk = c * 4 + half * 2;
#pragma unroll
      for (int j = 0; j < 8; ++j) {
        v2f bf = *(const v2f*)&kbuf[(j * 16 + l16) * LS + kk];
        s_acc[j] = WMMA_F32(qf[c], bf, s_acc[j]);
      }
    }
    __syncthreads(); // everyone done reading kbuf before P overwrites it

    // online softmax in registers (row = v + 8*half, cols across 16-lane group)
#pragma unroll
    for (int v = 0; v < 8; ++v) {
      float lm = -INFINITY;
#pragma unroll
      for (int j = 0; j < 8; ++j) lm = fmaxf(lm, s_acc[j][v]);
      lm = fmaxf(lm, __shfl_xor(lm, 1, 32));
      lm = fmaxf(lm, __shfl_xor(lm, 2, 32));
      lm = fmaxf(lm, __shfl_xor(lm, 4, 32));
      lm = fmaxf(lm, __shfl_xor(lm, 8, 32));
      float mnew  = fmaxf(m_i[v], lm);
      float alpha = __expf(m_i[v] - mnew);
      m_i[v] = mnew;
      float rs = 0.f;
#pragma unroll
      for (int j = 0; j < 8; ++j) {
        float p = __expf(s_acc[j][v] - mnew);
        s_acc[j][v] = p;
        rs += p;
      }
      rs += __shfl_xor(rs, 1, 32);
      rs += __shfl_xor(rs, 2, 32);
      rs += __shfl_xor(rs, 4, 32);
      rs += __shfl_xor(rs, 8, 32);
      l_i[v] = l_i[v] * alpha + rs;
#pragma unroll
      for (int j = 0; j < 8; ++j) o_acc[j][v] *= alpha;
    }

    // stage P into kbuf (per-wave row slice; LDS ops are in-order per wave)
#pragma unroll
    for (int j = 0; j < 8; ++j)
#pragma unroll
      for (int v = 0; v < 8; ++v)
        kbuf[(wave * 16 + v + 8 * half) * LS + j * 16 + l16] = s_acc[j][v];

    // O += P @ V
#pragma unroll 4
    for (int c = 0; c < 32; ++c) {
      const int kk = c * 4 + half * 2;
      v2f pf = *(const v2f*)&kbuf[(wave * 16 + l16) * LS + kk];
#pragma unroll
      for (int j = 0; j < 8; ++j) {
        v2f bv = *(const v2f*)&vbuf[(j * 16 + l16) * LS + kk];
        o_acc[j] = WMMA_F32(pf, bv, o_acc[j]);
      }
    }
    __syncthreads(); // before next tile overwrites kbuf/vbuf
  }

  // epilogue: normalize and write context [row=(b,s), col=(h,d)]
#pragma unroll
  for (int v = 0; v < 8; ++v) {
    float inv = 1.0f / l_i[v];
    size_t grow = (size_t)(b * S + q0 + v + 8 * half);
    float* crow = Ctx + grow * 2048 + h * 128;
#pragma unroll
    for (int j = 0; j < 8; ++j)
      crow[j * 16 + l16] = o_acc[j][v] * inv;
  }
}

// ---------------------------------------------------------------------------
extern "C" void kernel_launch(void* const* d_in, const int* in_sizes, int n_in,
                              void* d_out, int out_size, void* d_ws, size_t ws_size,
                              hipStream_t stream) {
  (void)in_sizes; (void)n_in; (void)out_size; (void)ws_size;
  const float* X  = (const float*)d_in[0];
  const float* Wq = (const float*)d_in[1];
  const float* Wk = (const float*)d_in[2];
  const float* Wv = (const float*)d_in[3];
  const float* Wo = (const float*)d_in[4];

  const int B = 2, S = 2048, H = 2048, NH = 16, HD = 128;
  const int M = B * S; // 4096

  float* out      = (float*)d_out;
  float* attn_out = out;                        // [M, H]
  float* k_out    = out + (size_t)M * H;        // [M, HD] (post-RoPE K)
  float* v_out    = k_out + (size_t)M * HD;     // [M, HD]

  float* ws    = (float*)d_ws;
  float* q_raw = ws;                            // [M, H]   (Q, RoPE in place)
  float* ctx   = q_raw + (size_t)M * H;         // [M, H]   (attention context)
  float* k_raw = ctx + (size_t)M * H;           // [M, HD]  (pre-RoPE K)

  dim3 blk(256);

  // projections
  gemm_xwt_f32<<<dim3(H / 128, M / 128), blk, 0, stream>>>(X, Wq, q_raw, M, H, H);
  gemm_xwt_f32<<<dim3(HD / 128, M / 128), blk, 0, stream>>>(X, Wk, k_raw, M, HD, H);
  gemm_xwt_f32<<<dim3(HD / 128, M / 128), blk, 0, stream>>>(X, Wv, v_out, M, HD, H);

  // rotary embedding
  rope_f32<<<dim3((M * NH * 64) / 256), blk, 0, stream>>>(q_raw, q_raw, M, NH, S);
  rope_f32<<<dim3((M * 64) / 256), blk, 0, stream>>>(k_raw, k_out, M, 1, S);

  // attention (flash, online softmax)
  mqa_attn_f32<<<dim3(B * NH * (S / 128)), blk, 0, stream>>>(q_raw, k_out, v_out, ctx, B, S);

  // output projection
  gemm_xwt_f32<<<dim3(H / 128, M / 128), blk, 0, stream>>>(ctx, Wo, attn_out, M, H, H);
}